// GraphSAGE_3315714752647
// MI455X (gfx1250) — compile-verified
//
#include <hip/hip_runtime.h>

#define NN 50000
#define NE 625000
#define DIM 128

typedef __attribute__((ext_vector_type(2))) float v2f;
typedef __attribute__((ext_vector_type(8))) float v8f;

// ---------------- utility: zero a float buffer ----------------
__global__ void sage_zero_kernel(float* __restrict__ p, long n) {
  long i = (long)blockIdx.x * blockDim.x + threadIdx.x;
  long stride = (long)gridDim.x * blockDim.x;
  for (; i < n; i += stride) p[i] = 0.0f;
}

// ---------------- in-degree (float counts) ----------------
__global__ void sage_degree_kernel(const int* __restrict__ dst,
                                   float* __restrict__ deg, int E) {
  int e = blockIdx.x * blockDim.x + threadIdx.x;
  if (e < E) atomicAdd(&deg[dst[e]], 1.0f);
}

// ---------------- edge scatter: agg[dst] += w * h[src] ----------------
// one wave32 per edge, each lane handles 4 features (float4)
__global__ __launch_bounds__(256) void sage_scatter_kernel(
    const float* __restrict__ h, const int* __restrict__ src,
    const int* __restrict__ dst, const float* __restrict__ w,
    float* __restrict__ agg, int E) {
  int t = blockIdx.x * blockDim.x + threadIdx.x;
  int lane = t & 31;
  int e = t >> 5;
  if (e >= E) return;
  int s = src[e];
  int d = dst[e];
  float we = w[e];
  const float4* row = (const float4*)(h + (size_t)s * DIM);
  float4 v = row[lane];                    // 32 lanes x 16B = full 512B row
  float* ap = agg + (size_t)d * DIM + lane * 4;
  atomicAdd(ap + 0, v.x * we);
  atomicAdd(ap + 1, v.y * we);
  atomicAdd(ap + 2, v.z * we);
  atomicAdd(ap + 3, v.w * we);
}

// ---------------- fused SAGE GEMM with f32 WMMA ----------------
// out[n,:] = act( h[n,:] @ Wself^T + (agg[n,:]/max(deg,1)) @ Wneigh^T + b )
#define RPB 128   // rows per block
#define KC  16    // K chunk
#define AP  20    // padded A row stride   (20*m mod 64 distinct, m=0..15)
#define WP  132   // padded Wt row stride  (4*k + j conflict-free)

__global__ __launch_bounds__(256) void sage_gemm_kernel(
    const float* __restrict__ Hself, const float* __restrict__ Agg,
    const float* __restrict__ Deg,   const float* __restrict__ Wself,
    const float* __restrict__ Wneigh,const float* __restrict__ Bias,
    float* __restrict__ Out, int nrows, int do_relu) {
  __shared__ float sWs[KC * WP];
  __shared__ float sWn[KC * WP];
  __shared__ float sAs[RPB * AP];
  __shared__ float sAn[RPB * AP];
  __shared__ float sRd[RPB];

  const int tid  = threadIdx.x;
  const int lane = tid & 31;
  const int wave = tid >> 5;                 // 0..7 -> 16-col stripe
  const int rowbase = blockIdx.x * RPB;

  // reciprocal degree for the block's rows
  for (int r = tid; r < RPB; r += 256) {
    int gr = rowbase + r;
    float dg = (gr < nrows) ? Deg[gr] : 1.0f;
    sRd[r] = 1.0f / fmaxf(dg, 1.0f);
  }

  v8f acc[8] = {};                           // 8 row tiles x 16x16 f32

  const int jj    = wave * 16 + (lane & 15); // output column of this lane
  const int khalf = (lane >> 4) << 1;        // K offset for upper half-lanes
  const int m     = lane & 15;               // A-matrix row of this lane

  for (int kc = 0; kc < DIM / KC; ++kc) {
    __syncthreads();
    // stage transposed W chunks: Wt[k][j] = W[j][kc*KC + k]
#pragma unroll
    for (int s = 0; s < (RPB * KC) / 256; ++s) {
      int i = tid + s * 256;
      int j = i >> 4;                        // 0..127
      int k = i & 15;                        // 0..15
      sWs[k * WP + j] = Wself [j * DIM + kc * KC + k];
      sWn[k * WP + j] = Wneigh[j * DIM + kc * KC + k];
    }
    // stage A chunks (self rows, and degree-normalized neighbor rows)
#pragma unroll
    for (int s = 0; s < (RPB * KC) / 256; ++s) {
      int i = tid + s * 256;
      int r = i >> 4;
      int k = i & 15;
      int gr = rowbase + r;
      float hs = 0.0f, hn = 0.0f;
      if (gr < nrows) {
        hs = Hself[(size_t)gr * DIM + kc * KC + k];
        hn = Agg  [(size_t)gr * DIM + kc * KC + k] * sRd[r];
      }
      sAs[r * AP + k] = hs;
      sAn[r * AP + k] = hn;
    }
    __syncthreads();

    // B fragments for this K chunk (shared by all 8 row tiles)
    v2f bs[4], bn[4];
#pragma unroll
    for (int k0 = 0; k0 < 4; ++k0) {
      int kk = k0 * 4 + khalf;               // {0,4,8,12} + {0,2}
      bs[k0].x = sWs[ kk      * WP + jj];
      bs[k0].y = sWs[(kk + 1) * WP + jj];
      bn[k0].x = sWn[ kk      * WP + jj];
      bn[k0].y = sWn[(kk + 1) * WP + jj];
    }

#pragma unroll
    for (int rt = 0; rt < 8; ++rt) {
      const float* as = &sAs[(rt * 16 + m) * AP];
      const float* an = &sAn[(rt * 16 + m) * AP];
#pragma unroll
      for (int k0 = 0; k0 < 4; ++k0) {
        int kk = k0 * 4 + khalf;
        v2f a_s; a_s.x = as[kk]; a_s.y = as[kk + 1];
        v2f a_n; a_n.x = an[kk]; a_n.y = an[kk + 1];
        acc[rt] = __builtin_amdgcn_wmma_f32_16x16x4_f32(
            false, a_s, false, bs[k0], (short)0, acc[rt], false, false);
        acc[rt] = __builtin_amdgcn_wmma_f32_16x16x4_f32(
            false, a_n, false, bn[k0], (short)0, acc[rt], false, false);
      }
    }
  }

  // epilogue: bias (+ReLU), guarded store. D layout: VGPR r -> M=r (+8 upper half)
  const float bv = Bias[jj];
  const int mofs = (lane >> 4) << 3;
#pragma unroll
  for (int rt = 0; rt < 8; ++rt) {
#pragma unroll
    for (int r = 0; r < 8; ++r) {
      int row = rowbase + rt * 16 + r + mofs;
      if (row < nrows) {
        float v = acc[rt][r] + bv;
        if (do_relu) v = fmaxf(v, 0.0f);
        Out[(size_t)row * DIM + jj] = v;
      }
    }
  }
}

// ---------------- launcher ----------------
extern "C" void kernel_launch(void* const* d_in, const int* in_sizes, int n_in,
                              void* d_out, int out_size, void* d_ws, size_t ws_size,
                              hipStream_t stream) {
  const float* x       = (const float*)d_in[0];
  const int*   src     = (const int*)  d_in[1];
  const int*   dst     = (const int*)  d_in[2];
  const float* weight  = (const float*)d_in[3];
  const float* Wself1  = (const float*)d_in[4];
  const float* Wneigh1 = (const float*)d_in[5];
  const float* b1      = (const float*)d_in[6];
  const float* Wself2  = (const float*)d_in[7];
  const float* Wneigh2 = (const float*)d_in[8];
  const float* b2      = (const float*)d_in[9];
  float* out = (float*)d_out;

  // workspace layout (bytes): deg[N] | agg[N*D] | h1[N*D]
  char* ws = (char*)d_ws;
  float* deg = (float*)(ws);
  float* agg = (float*)(ws + ((size_t)NN * 4 + 255 & ~(size_t)255));
  float* h1  = (float*)((char*)agg + (size_t)NN * DIM * 4);

  const long aggN = (long)NN * DIM;
  const int gemm_blocks = (NN + RPB - 1) / RPB;            // 391
  const int scat_blocks = (NE * 32 + 255) / 256;           // wave per edge
  const int deg_blocks  = (NE + 255) / 256;

  // ---- layer 1 ----
  sage_zero_kernel<<<2048, 256, 0, stream>>>(deg, NN);
  sage_zero_kernel<<<2048, 256, 0, stream>>>(agg, aggN);
  sage_degree_kernel<<<deg_blocks, 256, 0, stream>>>(dst, deg, NE);
  sage_scatter_kernel<<<scat_blocks, 256, 0, stream>>>(x, src, dst, weight, agg, NE);
  sage_gemm_kernel<<<gemm_blocks, 256, 0, stream>>>(
      x, agg, deg, Wself1, Wneigh1, b1, h1, NN, /*relu=*/1);

  // ---- layer 2 (degree unchanged) ----
  sage_zero_kernel<<<2048, 256, 0, stream>>>(agg, aggN);
  sage_scatter_kernel<<<scat_blocks, 256, 0, stream>>>(h1, src, dst, weight, agg, NE);
  sage_gemm_kernel<<<gemm_blocks, 256, 0, stream>>>(
      h1, agg, deg, Wself2, Wneigh2, b2, out, NN, /*relu=*/0);
}